// Refine_39831526703901
// MI455X (gfx1250) — compile-verified
//
#include <hip/hip_runtime.h>
#include <stdint.h>

// Problem constants (from reference): X,G1,G2,G3 are [B,C,H,W] fp32
#define NB 4
#define NC 32
#define NH 512
#define NW 512

#define TT 8                 // columns staged per tile
#define NT (NW / TT)         // 64 tiles
#define STRIDE 9             // dwords per LDS tile row (odd -> conflict-free banks)
#define NARR 4               // X, G1, G2, G3

typedef float float4v __attribute__((ext_vector_type(4)));

#define AS3 __attribute__((address_space(3)))
#define AS1 __attribute__((address_space(1)))

// ---- CDNA5 async global->LDS copy (ASYNCcnt-tracked) -----------------------
__device__ __forceinline__ void async_cp_b32(AS3 float* lds, AS1 float* gsrc) {
#if __has_builtin(__builtin_amdgcn_global_load_async_to_lds_b32)
    __builtin_amdgcn_global_load_async_to_lds_b32((AS1 int*)gsrc, (AS3 int*)lds, 0, 0);
#else
    unsigned loff = (unsigned)(uintptr_t)lds;   // LDS byte offset
    asm volatile("global_load_async_to_lds_b32 %0, %1, off"
                 :: "v"(loff), "v"(gsrc)
                 : "memory");
#endif
}

#if __has_builtin(__builtin_amdgcn_s_wait_asynccnt)
#define WAIT_ASYNC(N) __builtin_amdgcn_s_wait_asynccnt(N)
#else
#define WAIT_ASYNC(N) asm volatile("s_wait_asynccnt " #N ::: "memory")
#endif

// Issue async loads for one tile: lane h copies TT contiguous floats of its
// own row, for each of the 4 input arrays, into LDS buffer `buf`.
__device__ __forceinline__ void issue_tile_loads(
    float tiles[2][NARR][NH * STRIDE], int buf,
    const float* s0, const float* s1, const float* s2, const float* s3,
    int t0, int h)
{
    const float* srcs[NARR] = {s0, s1, s2, s3};
#pragma unroll
    for (int a = 0; a < NARR; ++a) {
        AS3 float* drow = (AS3 float*)&tiles[buf][a][h * STRIDE];
        AS1 float* srow = (AS1 float*)(srcs[a] + t0);
#pragma unroll
        for (int j = 0; j < TT; ++j) {
            async_cp_b32(drow + j, srow + j);
        }
    }
}

__global__ __launch_bounds__(NH, 1)
void spn_rl_scan_kernel(const float* __restrict__ X,
                        const float* __restrict__ G1,
                        const float* __restrict__ G2,
                        const float* __restrict__ G3,
                        float* __restrict__ OUT)
{
    // Double-buffered staging tiles: 2 * 4 * 512 * 9 * 4B = 147,456 B
    __shared__ float tiles[2][NARR][NH * STRIDE];
    // Double-buffered h exchange with zero halos: 2 * 514 * 4B = 4,112 B
    __shared__ float hbuf[2][NH + 2];

    const int h     = threadIdx.x;        // row owned by this lane
    const int slice = blockIdx.x;         // (b*C + c)
    const size_t rowbase = ((size_t)slice * NH + (size_t)h) * NW;

    const float* xr  = X  + rowbase;
    const float* g1r = G1 + rowbase;
    const float* g2r = G2 + rowbase;
    const float* g3r = G3 + rowbase;

    // init hidden-state exchange buffers (h(:, W) == 0, halos == 0 forever)
    hbuf[0][h + 1] = 0.0f;
    if (h < 2) {
        hbuf[0][h * (NH + 1)] = 0.0f;     // hbuf[0][0], hbuf[0][513]
        hbuf[1][h * (NH + 1)] = 0.0f;     // hbuf[1][0], hbuf[1][513]
    }

    // Prologue: prefetch the two right-most tiles.
    issue_tile_loads(tiles, (NT - 1) & 1, xr, g1r, g2r, g3r, (NT - 1) * TT, h);
    issue_tile_loads(tiles, (NT - 2) & 1, xr, g1r, g2r, g3r, (NT - 2) * TT, h);

    __syncthreads();                       // hbuf init visible to all waves

    int cur = 0;
    for (int tile = NT - 1; tile >= 0; --tile) {
        // Tile data is produced and consumed by the same lane -> per-wave wait
        // suffices. >32 outstanding means the current tile's 32 ops are the
        // older ones; waiting to <=32 guarantees they completed (in-order done).
        if (tile > 0) { WAIT_ASYNC(32); } else { WAIT_ASYNC(0); }

        const int pb = tile & 1;
        const int t0 = tile * TT;
        float outv[TT];

#pragma unroll
        for (int tt = TT - 1; tt >= 0; --tt) {
            const int idx = h * STRIDE + tt;
            float x  = tiles[pb][0][idx];
            float g1 = tiles[pb][1][idx];
            float g2 = tiles[pb][2][idx];
            float g3 = tiles[pb][3][idx];

            // gate normalization: rescale iff |g1|+|g2|+|g3| >= 1
            float s = fabsf(g1) + fabsf(g2) + fabsf(g3);
            float scale = (s >= 1.0f) ? (1.0f / s) : 1.0f;
            g1 *= scale; g2 *= scale; g3 *= scale;

            float hu = hbuf[cur][h];        // h(row-1, t+1)
            float hm = hbuf[cur][h + 1];    // h(row,   t+1)
            float hd = hbuf[cur][h + 2];    // h(row+1, t+1)

            float hn = (1.0f - g1 - g2 - g3) * x + g1 * hu + g2 * hm + g3 * hd;

            hbuf[cur ^ 1][h + 1] = hn;
            outv[tt] = hn;

            __syncthreads();                // publish hn for next column step
            cur ^= 1;
        }

        // Flush this tile's outputs: two aligned 16B stores per row.
        float4v v0 = {outv[0], outv[1], outv[2], outv[3]};
        float4v v1 = {outv[4], outv[5], outv[6], outv[7]};
        *(float4v*)(OUT + rowbase + t0)     = v0;
        *(float4v*)(OUT + rowbase + t0 + 4) = v1;

        // Buffer pb is now free: prefetch tile-2 (same parity) behind compute.
        if (tile >= 2) {
            issue_tile_loads(tiles, pb, xr, g1r, g2r, g3r, (tile - 2) * TT, h);
        }
    }
}

extern "C" void kernel_launch(void* const* d_in, const int* in_sizes, int n_in,
                              void* d_out, int out_size, void* d_ws, size_t ws_size,
                              hipStream_t stream) {
    (void)in_sizes; (void)n_in; (void)out_size; (void)d_ws; (void)ws_size;
    const float* X  = (const float*)d_in[0];
    const float* G1 = (const float*)d_in[1];
    const float* G2 = (const float*)d_in[2];
    const float* G3 = (const float*)d_in[3];
    float* OUT = (float*)d_out;

    dim3 grid(NB * NC);   // 128 workgroups, one per (b,c) slice
    dim3 block(NH);       // 512 threads = 16 wave32, lane h owns row h

    hipLaunchKernelGGL(spn_rl_scan_kernel, grid, block, 0, stream,
                       X, G1, G2, G3, OUT);
}